// DynamicVoxelizer_31550829757172
// MI455X (gfx1250) — compile-verified
//
#include <hip/hip_runtime.h>

typedef float v4f __attribute__((ext_vector_type(4)));

// Voxelizer constants (from reference)
#define VSX 0.1f
#define VSY 0.1f
#define VSZ 0.2f
#define PMINX (-51.2f)
#define PMINY (-51.2f)
#define PMINZ (-5.0f)
#define GRIDX 1024
#define GRIDY 1024
#define GRIDZ 40
#define BLOCK 256

__global__ __launch_bounds__(BLOCK) void DynamicVoxelizer_31550829757172_kernel(
    const float* __restrict__ pts,   // [B*N*4]
    float* __restrict__ o_points,    // [B*N*4]
    float* __restrict__ o_coords,    // [B*N*3]  (z,y,x) or -1
    float* __restrict__ o_idx,       // [B*N]    point index in batch or -1
    float* __restrict__ o_offs,      // [B*N*3]  xyz - center or 0
    float* __restrict__ o_valid,     // [B*N]    1.0 / 0.0
    int N)
{
    __shared__ float s_coords[BLOCK * 3];
    __shared__ float s_offs[BLOCK * 3];

    const int t = threadIdx.x;
    const int b = blockIdx.y;
    const int base_in_batch = blockIdx.x * BLOCK;
    const int i_in_batch = base_in_batch + t;
    const bool active = i_in_batch < N;
    const long long idx = (long long)b * N + i_in_batch;

    v4f p = (v4f)0.0f;
    if (active) {
        p = __builtin_nontemporal_load((const v4f*)pts + idx);
    }
    const float x = p.x, y = p.y, z = p.z, w = p.w;

    const bool anynan = __builtin_isnan(x) | __builtin_isnan(y) |
                        __builtin_isnan(z) | __builtin_isnan(w);

    // cf = floor((xyz - pmin) / vs)  -- IEEE f32 divide to match jnp bit-exactly
    const float cfx = __builtin_floorf((x - PMINX) / VSX);
    const float cfy = __builtin_floorf((y - PMINY) / VSY);
    const float cfz = __builtin_floorf((z - PMINZ) / VSZ);
    const int cx = (int)cfx;
    const int cy = (int)cfy;
    const int cz = (int)cfz;

    const bool in_range = (cx >= 0) & (cx < GRIDX) &
                          (cy >= 0) & (cy < GRIDY) &
                          (cz >= 0) & (cz < GRIDZ);
    const bool valid = (!anynan) & in_range;

    // centers = cf*vs + pmin + vs*0.5 ; offsets gated by valid
    const float cenx = cfx * VSX + PMINX + 0.5f * VSX;
    const float ceny = cfy * VSY + PMINY + 0.5f * VSY;
    const float cenz = cfz * VSZ + PMINZ + 0.5f * VSZ;
    const float offx = valid ? (x - cenx) : 0.0f;
    const float offy = valid ? (y - ceny) : 0.0f;
    const float offz = valid ? (z - cenz) : 0.0f;

    // ---- directly coalesced outputs: NT stores (write-once streams) ----
    if (active) {
        v4f po = valid ? p : (v4f)0.0f;
        __builtin_nontemporal_store(po, (v4f*)o_points + idx);
        __builtin_nontemporal_store(valid ? (float)i_in_batch : -1.0f, o_idx + idx);
        __builtin_nontemporal_store(valid ? 1.0f : 0.0f, o_valid + idx);
    }

    // ---- stride-3 outputs: stage in LDS, then contiguous b128 NT stores ----
    // stride-3 dword writes: gcd(3,64)=1 -> bank-conflict free
    s_coords[t * 3 + 0] = valid ? (float)cz : -1.0f;   // reversed: z,y,x
    s_coords[t * 3 + 1] = valid ? (float)cy : -1.0f;
    s_coords[t * 3 + 2] = valid ? (float)cx : -1.0f;
    s_offs[t * 3 + 0] = offx;
    s_offs[t * 3 + 1] = offy;
    s_offs[t * 3 + 2] = offz;

    __syncthreads();  // s_wait_dscnt + CDNA5 split barrier

    const int nblk = (N - base_in_batch) < BLOCK ? (N - base_in_batch) : BLOCK;
    const int nfloat = nblk * 3;
    const int nvec = nfloat >> 2;
    const long long gbase = ((long long)b * N + base_in_batch) * 3;  // 16B-aligned

    const v4f* sc4 = (const v4f*)s_coords;
    const v4f* so4 = (const v4f*)s_offs;
    v4f* gc4 = (v4f*)(o_coords + gbase);
    v4f* go4 = (v4f*)(o_offs + gbase);

    for (int j = t; j < nvec; j += BLOCK) {
        __builtin_nontemporal_store(sc4[j], gc4 + j);
        __builtin_nontemporal_store(so4[j], go4 + j);
    }
    // scalar tail (only in the last partial block of each batch)
    for (int j = (nvec << 2) + t; j < nfloat; j += BLOCK) {
        __builtin_nontemporal_store(s_coords[j], o_coords + gbase + j);
        __builtin_nontemporal_store(s_offs[j], o_offs + gbase + j);
    }
}

extern "C" void kernel_launch(void* const* d_in, const int* in_sizes, int n_in,
                              void* d_out, int out_size, void* d_ws, size_t ws_size,
                              hipStream_t stream) {
    const float* pts = (const float*)d_in[0];
    const int C = 4;
    const int N = 1000000;                       // fixed by reference
    const long long BN = (long long)in_sizes[0] / C;  // B*N
    const int B = (int)(BN / N);

    float* out = (float*)d_out;
    float* o_points = out;                 // BN*4
    float* o_coords = out + BN * 4;        // BN*3
    float* o_idx    = out + BN * 7;        // BN
    float* o_offs   = out + BN * 8;        // BN*3
    float* o_valid  = out + BN * 11;       // BN

    dim3 grid((N + BLOCK - 1) / BLOCK, B);
    dim3 block(BLOCK);
    hipLaunchKernelGGL(DynamicVoxelizer_31550829757172_kernel, grid, block, 0, stream,
                       pts, o_points, o_coords, o_idx, o_offs, o_valid, N);
}